// EdgeAttentionGAT_79637283602847
// MI455X (gfx1250) — compile-verified
//
#include <hip/hip_runtime.h>
#include <hip/hip_bf16.h>
#include <stdint.h>

typedef float v2f __attribute__((ext_vector_type(2)));
typedef float v8f __attribute__((ext_vector_type(8)));

// ---------------- helpers ----------------
__device__ __forceinline__ float atomAddF(float* p, float v) {
    return __hip_atomic_fetch_add(p, v, __ATOMIC_RELAXED, __HIP_MEMORY_SCOPE_AGENT);
}
// monotonic float->uint key for atomicMax over floats
__device__ __forceinline__ unsigned fkey(float f) {
    unsigned u = __float_as_uint(f);
    return (u & 0x80000000u) ? ~u : (u | 0x80000000u);
}
__device__ __forceinline__ float funkey(unsigned k) {
    unsigned u = (k & 0x80000000u) ? (k & 0x7FFFFFFFu) : ~k;
    return __uint_as_float(u);
}
__device__ __forceinline__ float silu(float x) { return x / (1.0f + __expf(-x)); }

// ---------------- zero init (grid-stride) ----------------
__global__ void eg_zero_kernel(float* __restrict__ p, size_t n) {
    size_t i = (size_t)blockIdx.x * blockDim.x + threadIdx.x;
    size_t stride = (size_t)gridDim.x * blockDim.x;
    for (; i < n; i += stride) p[i] = 0.0f;
}

// ---------------- edge MLP: e = silu(ef@W1+b1)@W2+b2  [E,3] ----------------
__global__ void eg_edge_mlp_kernel(const float* __restrict__ ef,
                                   const float* __restrict__ W1, const float* __restrict__ b1,
                                   const float* __restrict__ W2, const float* __restrict__ b2,
                                   float* __restrict__ e, int E) {
    int i = blockIdx.x * blockDim.x + threadIdx.x;
    if (i >= E) return;
    float x0 = ef[i * 3 + 0], x1 = ef[i * 3 + 1], x2 = ef[i * 3 + 2];
    float hmid[6];
#pragma unroll
    for (int j = 0; j < 6; ++j) {
        float s = x0 * W1[j] + x1 * W1[6 + j] + x2 * W1[12 + j] + b1[j];
        hmid[j] = silu(s);
    }
#pragma unroll
    for (int j = 0; j < 3; ++j) {
        float s = b2[j];
#pragma unroll
        for (int k = 0; k < 6; ++k) s += hmid[k] * W2[k * 3 + j];
        e[i * 3 + j] = s;
    }
}

// ---------------- fp32 WMMA GEMM: C[nrows,128] = A[nrows,128] @ B[128,128] (+bias) ----
// 8 waves / block; wave w handles the 16-col tile w; blockIdx.x = 16-row tile.
__global__ __launch_bounds__(256) void eg_gemm128_wmma(const float* __restrict__ A,
                                                       const float* __restrict__ B,
                                                       const float* __restrict__ bias,
                                                       float* __restrict__ C) {
    const int wave = threadIdx.x >> 5;
    const int lane = threadIdx.x & 31;
    const int r0 = blockIdx.x * 16;
    const int c0 = wave * 16;
    const int mn = lane & 15;          // A row within tile / C,D column within tile
    const int kk = (lane >> 4) * 2;    // K sub-offset for this lane half
    const float* Ap = A + (size_t)(r0 + mn) * 128 + kk;
    const float* Bp = B + (size_t)kk * 128 + (c0 + mn);
    v8f acc = {};
#pragma unroll 4
    for (int k = 0; k < 128; k += 4) {
        v2f a = *(const v2f*)(Ap + k);                 // A[m][k+kk], A[m][k+kk+1]
        v2f b;
        b.x = Bp[(size_t)k * 128];                     // B[k+kk][n]
        b.y = Bp[(size_t)(k + 1) * 128];               // B[k+kk+1][n]
        acc = __builtin_amdgcn_wmma_f32_16x16x4_f32(false, a, false, b,
                                                    (short)0, acc, false, false);
    }
    const float bv = bias ? bias[c0 + mn] : 0.0f;      // per-column bias
    float* Cp = C + (size_t)r0 * 128 + c0 + mn;
    const int mbase = (lane < 16) ? 0 : 8;
#pragma unroll
    for (int i = 0; i < 8; ++i)
        Cp[(size_t)(mbase + i) * 128] = acc[i] + bv;
}

// scalar tail for nrows not multiple of 16 (not used for N=50000, kept for safety)
__global__ void eg_gemm128_tail(const float* __restrict__ A, const float* __restrict__ B,
                                const float* __restrict__ bias, float* __restrict__ C,
                                int rowStart, int nrows) {
    int total = (nrows - rowStart) * 128;
    int idx = blockIdx.x * blockDim.x + threadIdx.x;
    if (idx >= total) return;
    int r = rowStart + idx / 128, c = idx % 128;
    float s = bias ? bias[c] : 0.0f;
    for (int k = 0; k < 128; ++k) s += A[(size_t)r * 128 + k] * B[(size_t)k * 128 + c];
    C[(size_t)r * 128 + c] = s;
}

// ---------------- degree + edge-feature scatter (replaces concat scatter) ---------
__global__ void eg_deg_esum_kernel(const int* __restrict__ src, const int* __restrict__ dst,
                                   const float* __restrict__ e,
                                   float* __restrict__ deg, float* __restrict__ esum, int E) {
    int i = blockIdx.x * blockDim.x + threadIdx.x;
    if (i >= E) return;
    int s = src[i], d = dst[i];
    atomAddF(&deg[s], 1.0f);
    atomAddF(&deg[d], 1.0f);
#pragma unroll
    for (int k = 0; k < 3; ++k) {
        float ev = e[i * 3 + k];
        atomAddF(&esum[s * 3 + k], ev);
        atomAddF(&esum[d * 3 + k], ev);
    }
}

// ---------------- FC epilogue: h = deg*q + esum@Wfc[128:131]; el/er dots ----------
// one wave per node; lane owns 4 contiguous features (head = lane/8)
__global__ void eg_fc_post_kernel(float* __restrict__ h, const float* __restrict__ deg,
                                  const float* __restrict__ esum, const float* __restrict__ Wfc,
                                  const float* __restrict__ attn_l, const float* __restrict__ attn_r,
                                  float* __restrict__ el, float* __restrict__ er, int n) {
    int wave = threadIdx.x >> 5, lane = threadIdx.x & 31;
    int v = blockIdx.x * 8 + wave;
    if (v >= n) return;
    float d  = deg[v];
    float e0 = esum[v * 3 + 0], e1 = esum[v * 3 + 1], e2 = esum[v * 3 + 2];
    int j = lane * 4;
    int head = j >> 5;
    int f0 = j & 31;
    float4 q = *(const float4*)(h + (size_t)v * 128 + j);
    const float* w0 = Wfc + (size_t)128 * 128;
    const float* w1 = Wfc + (size_t)129 * 128;
    const float* w2 = Wfc + (size_t)130 * 128;
    const float* al = attn_l + head * 32;
    const float* ar = attn_r + head * 32;
    float qa[4] = {q.x, q.y, q.z, q.w};
    float hv[4];
    float pl = 0.0f, pr = 0.0f;
#pragma unroll
    for (int t = 0; t < 4; ++t) {
        float val = d * qa[t] + e0 * w0[j + t] + e1 * w1[j + t] + e2 * w2[j + t];
        hv[t] = val;
        pl += val * al[f0 + t];
        pr += val * ar[f0 + t];
    }
    *(float4*)(h + (size_t)v * 128 + j) = make_float4(hv[0], hv[1], hv[2], hv[3]);
#pragma unroll
    for (int off = 1; off < 8; off <<= 1) {
        pl += __shfl_xor(pl, off, 32);
        pr += __shfl_xor(pr, off, 32);
    }
    if ((lane & 7) == 0) {
        el[v * 4 + head] = pl;
        er[v * 4 + head] = pr;
    }
}

// ---------------- logits + segment max (keyed atomicMax) ----------------
__global__ void eg_logits_max_kernel(const int* __restrict__ src, const int* __restrict__ dst,
                                     const float* __restrict__ el, const float* __restrict__ er,
                                     float* __restrict__ a, unsigned* __restrict__ m, int E) {
    int i = blockIdx.x * blockDim.x + threadIdx.x;   // over E*4
    if (i >= E * 4) return;
    int e = i >> 2, hd = i & 3;
    float v = el[src[e] * 4 + hd] + er[dst[e] * 4 + hd];
    v = v > 0.0f ? v : 0.2f * v;                      // leaky_relu
    a[i] = v;
    atomicMax(&m[dst[e] * 4 + hd], fkey(v));
}

// ---------------- exp + segment sum ----------------
__global__ void eg_exp_sum_kernel(const int* __restrict__ dst, const unsigned* __restrict__ m,
                                  float* __restrict__ a, float* __restrict__ z, int E) {
    int i = blockIdx.x * blockDim.x + threadIdx.x;
    if (i >= E * 4) return;
    int e = i >> 2, hd = i & 3;
    float mx = funkey(m[dst[e] * 4 + hd]);
    float av = __expf(a[i] - mx);
    a[i] = av;
    atomAddF(&z[dst[e] * 4 + hd], av);
}

// ---------------- attention aggregation: rst[dst] += alpha * h[src] --------------
// one wave per edge; lane owns 4 contiguous features of its head
__global__ void eg_aggregate_kernel(const int* __restrict__ src, const int* __restrict__ dst,
                                    const float* __restrict__ a, const float* __restrict__ z,
                                    const float* __restrict__ h, float* __restrict__ rst, int E) {
    int wave = threadIdx.x >> 5, lane = threadIdx.x & 31;
    int e = blockIdx.x * 8 + wave;
    if (e >= E) return;
    int s = src[e], d = dst[e];
    int j = lane * 4;
    int head = j >> 5;
    float alpha = a[e * 4 + head] / z[d * 4 + head];
    float4 hs = *(const float4*)(h + (size_t)s * 128 + j);
    float* r = rst + (size_t)d * 128 + j;
    atomAddF(r + 0, alpha * hs.x);
    atomAddF(r + 1, alpha * hs.y);
    atomAddF(r + 2, alpha * hs.z);
    atomAddF(r + 3, alpha * hs.w);
}

// ---------------- residual + LayerNorm + SiLU ----------------
__global__ void eg_finalize_kernel(const float* __restrict__ rst, const float* __restrict__ nf,
                                   const float* __restrict__ gbias, const float* __restrict__ gamma,
                                   const float* __restrict__ beta, float* __restrict__ out, int n) {
    int wave = threadIdx.x >> 5, lane = threadIdx.x & 31;
    int v = blockIdx.x * 8 + wave;
    if (v >= n) return;
    int j = lane * 4;
    float4 r = *(const float4*)(rst + (size_t)v * 128 + j);
    float4 x = *(const float4*)(nf + (size_t)v * 128 + j);
    float4 g = *(const float4*)(gbias + j);
    float o[4] = {r.x + g.x + x.x, r.y + g.y + x.y, r.z + g.z + x.z, r.w + g.w + x.w};
    float s = o[0] + o[1] + o[2] + o[3];
    float q = o[0] * o[0] + o[1] * o[1] + o[2] * o[2] + o[3] * o[3];
#pragma unroll
    for (int off = 1; off < 32; off <<= 1) {
        s += __shfl_xor(s, off, 32);
        q += __shfl_xor(q, off, 32);
    }
    float mu = s * (1.0f / 128.0f);
    float var = q * (1.0f / 128.0f) - mu * mu;
    float inv = rsqrtf(var + 1e-5f);
    float4 ga = *(const float4*)(gamma + j);
    float4 be = *(const float4*)(beta + j);
    float gaa[4] = {ga.x, ga.y, ga.z, ga.w};
    float bea[4] = {be.x, be.y, be.z, be.w};
    float res[4];
#pragma unroll
    for (int t = 0; t < 4; ++t) {
        float y = (o[t] - mu) * inv * gaa[t] + bea[t];
        res[t] = silu(y);
    }
    *(float4*)(out + (size_t)v * 128 + j) = make_float4(res[0], res[1], res[2], res[3]);
}

static inline size_t align4(size_t x) { return (x + 3) & ~(size_t)3; }

extern "C" void kernel_launch(void* const* d_in, const int* in_sizes, int n_in,
                              void* d_out, int out_size, void* d_ws, size_t ws_size,
                              hipStream_t stream) {
    const float* node_feats = (const float*)d_in[0];
    const float* edge_feat  = (const float*)d_in[1];
    const int*   src        = (const int*)d_in[2];
    const int*   dst        = (const int*)d_in[3];
    const float* W_em1      = (const float*)d_in[4];
    const float* b_em1      = (const float*)d_in[5];
    const float* W_em2      = (const float*)d_in[6];
    const float* b_em2      = (const float*)d_in[7];
    const float* W_np       = (const float*)d_in[8];
    const float* b_np       = (const float*)d_in[9];
    const float* W_fc       = (const float*)d_in[10];
    const float* attn_l     = (const float*)d_in[11];
    const float* attn_r     = (const float*)d_in[12];
    const float* gat_bias   = (const float*)d_in[13];
    const float* ln_gamma   = (const float*)d_in[14];
    const float* ln_beta    = (const float*)d_in[15];
    float* out = (float*)d_out;

    const int N = in_sizes[0] / 128;
    const int E = in_sizes[2];
    if (N <= 0 || E <= 0) return;
    const size_t N128 = (size_t)N * 128;

    // workspace layout (floats)
    float* ws = (float*)d_ws;
    size_t o_e   = 0;                                   // [E,3] edge MLP out
    size_t o_p   = align4(o_e + (size_t)E * 3);         // [N,128] node proj
    size_t o_h   = o_p + N128;                          // [N,128] q -> h (in place)
    size_t o_a   = o_h + N128;                          // [E,4] logits -> exp
    size_t o_el  = align4(o_a + (size_t)E * 4);         // [N,4]
    size_t o_er  = o_el + (size_t)N * 4;                // [N,4]
    // zeroed block starts here:
    size_t o_deg = o_er + (size_t)N * 4;                // [N]
    size_t o_esum= o_deg + (size_t)N;                   // [N,3]
    size_t o_m   = align4(o_esum + (size_t)N * 3);      // [N,4] (unsigned keys)
    size_t o_z   = o_m + (size_t)N * 4;                 // [N,4]
    size_t o_rst = align4(o_z + (size_t)N * 4);         // [N,128]
    size_t zeroCount = (o_rst + N128) - o_deg;

    float* e_buf  = ws + o_e;
    float* p_buf  = ws + o_p;
    float* h_buf  = ws + o_h;
    float* a_buf  = ws + o_a;
    float* el_buf = ws + o_el;
    float* er_buf = ws + o_er;
    float* deg    = ws + o_deg;
    float* esum   = ws + o_esum;
    unsigned* m_u = (unsigned*)(ws + o_m);
    float* z_buf  = ws + o_z;
    float* rst    = ws + o_rst;

    // 1) zero accumulators (deg, esum, m-keys, z, rst)
    {
        int blocks = (int)((zeroCount + 255) / 256);
        if (blocks > 4096) blocks = 4096;
        eg_zero_kernel<<<blocks, 256, 0, stream>>>(deg, zeroCount);
    }
    // 2) edge MLP
    eg_edge_mlp_kernel<<<(E + 255) / 256, 256, 0, stream>>>(edge_feat, W_em1, b_em1,
                                                            W_em2, b_em2, e_buf, E);
    // 3) p = node_feats @ W_np + b_np   (fp32 WMMA)
    {
        int tiles = N / 16;
        if (tiles > 0)
            eg_gemm128_wmma<<<tiles, 256, 0, stream>>>(node_feats, W_np, b_np, p_buf);
        int tail = N - tiles * 16;
        if (tail > 0)
            eg_gemm128_tail<<<(tail * 128 + 255) / 256, 256, 0, stream>>>(
                node_feats, W_np, b_np, p_buf, tiles * 16, N);
    }
    // 4) degree + edge-feature scatter
    eg_deg_esum_kernel<<<(E + 255) / 256, 256, 0, stream>>>(src, dst, e_buf, deg, esum, E);
    // 5) q = p @ W_fc[:128,:]           (fp32 WMMA, no bias)
    {
        int tiles = N / 16;
        if (tiles > 0)
            eg_gemm128_wmma<<<tiles, 256, 0, stream>>>(p_buf, W_fc, nullptr, h_buf);
        int tail = N - tiles * 16;
        if (tail > 0)
            eg_gemm128_tail<<<(tail * 128 + 255) / 256, 256, 0, stream>>>(
                p_buf, W_fc, nullptr, h_buf, tiles * 16, N);
    }
    // 6) h = deg*q + esum@W_fc[128:131]; attention dot products el/er
    eg_fc_post_kernel<<<(N + 7) / 8, 256, 0, stream>>>(h_buf, deg, esum, W_fc,
                                                       attn_l, attn_r, el_buf, er_buf, N);
    // 7) leaky-relu logits + segment max over dst
    eg_logits_max_kernel<<<(E * 4 + 255) / 256, 256, 0, stream>>>(src, dst, el_buf, er_buf,
                                                                  a_buf, m_u, E);
    // 8) exp(logit - max) + segment sum
    eg_exp_sum_kernel<<<(E * 4 + 255) / 256, 256, 0, stream>>>(dst, m_u, a_buf, z_buf, E);
    // 9) rst[dst] += alpha * h[src]
    eg_aggregate_kernel<<<(E + 7) / 8, 256, 0, stream>>>(src, dst, a_buf, z_buf, h_buf, rst, E);
    // 10) residual + LayerNorm + SiLU
    eg_finalize_kernel<<<(N + 7) / 8, 256, 0, stream>>>(rst, node_feats, gat_bias,
                                                        ln_gamma, ln_beta, out, N);
}